// CVQuantumLayer_16827681866300
// MI455X (gfx1250) — compile-verified
//
#include <hip/hip_runtime.h>

typedef float v2f __attribute__((ext_vector_type(2)));
typedef float v8f __attribute__((ext_vector_type(8)));

#define NMODES 16
#define NLAYERS 6

// ---------------------------------------------------------------------------
// Setup: collapse the whole circuit (both parameter sets) into a 16x32 affine
// map.  ws[0..511] = Wt[16 inputs][32 outputs], ws[512..543] = bias[32].
// ---------------------------------------------------------------------------
__global__ __launch_bounds__(192) void cv_setup_kernel(
    const float* __restrict__ disp,   // [2][L][N][2]
    const float* __restrict__ sq,     // [2][L][N][2]
    const float* __restrict__ bs,     // [2][L][N-1][2]
    float* __restrict__ ws)
{
  __shared__ float s_dx[2][NLAYERS][NMODES];
  __shared__ float s_dp[2][NLAYERS][NMODES];
  __shared__ float s_sa[2][NLAYERS][NMODES];      // ch - cp*sh
  __shared__ float s_sb[2][NLAYERS][NMODES];      // -sp*sh
  __shared__ float s_sd[2][NLAYERS][NMODES];      // ch + cp*sh
  __shared__ float s_ct[2][NLAYERS][NMODES - 1];  // cos(th)
  __shared__ float s_cs[2][NLAYERS][NMODES - 1];  // cos(bphi)*sin(th)
  __shared__ float s_ss[2][NLAYERS][NMODES - 1];  // sin(bphi)*sin(th)

  const int t = threadIdx.x;

  // Phase 1: parameter -> coefficient conversion, fully parallel.
  if (t < 2 * NLAYERS * NMODES) {
    int cc = t / (NLAYERS * NMODES);
    int r  = t % (NLAYERS * NMODES);
    int l  = r / NMODES, i = r % NMODES;
    int base = ((cc * NLAYERS + l) * NMODES + i) * 2;
    float a = disp[base + 0], dphi = disp[base + 1];
    s_dx[cc][l][i] = 2.0f * a * cosf(dphi);
    s_dp[cc][l][i] = 2.0f * a * sinf(dphi);
    float rr = fabsf(sq[base + 0]);
    float sphi = sq[base + 1];
    float ch = coshf(rr), sh = sinhf(rr);
    float cp = cosf(sphi), sp = sinf(sphi);
    s_sa[cc][l][i] = ch - cp * sh;
    s_sb[cc][l][i] = -sp * sh;
    s_sd[cc][l][i] = ch + cp * sh;
  }
  if (t < 2 * NLAYERS * (NMODES - 1)) {
    int cc = t / (NLAYERS * (NMODES - 1));
    int r  = t % (NLAYERS * (NMODES - 1));
    int l  = r / (NMODES - 1), w = r % (NMODES - 1);
    int base = ((cc * NLAYERS + l) * (NMODES - 1) + w) * 2;
    float th   = 1.0f / (1.0f + expf(-bs[base + 0]));
    float bphi = bs[base + 1];
    float ct = cosf(th), st = sinf(th);
    s_ct[cc][l][w] = ct;
    s_cs[cc][l][w] = cosf(bphi) * st;
    s_ss[cc][l][w] = sinf(bphi) * st;
  }
  __syncthreads();

  // Phase 2: 34 probe columns (2 circuits x (16 basis vectors + 1 bias run)).
  if (t >= 34) return;
  const int cc  = t / 17;
  const int col = t % 17;
  const bool use_disp = (col == 16);

  float mx[NMODES], mp[NMODES];
#pragma unroll
  for (int i = 0; i < NMODES; ++i) {
    mx[i] = (!use_disp && i == col) ? 2.0f : 0.0f;  // mx_init = SQRT_2HBAR * x
    mp[i] = 0.0f;
  }

  for (int l = 0; l < NLAYERS; ++l) {
#pragma unroll
    for (int i = 0; i < NMODES; ++i) {
      float xv = mx[i], pv = mp[i];
      if (use_disp) { xv += s_dx[cc][l][i]; pv += s_dp[cc][l][i]; }
      float A = s_sa[cc][l][i], Bb = s_sb[cc][l][i], D = s_sd[cc][l][i];
      mx[i] = A * xv + Bb * pv;
      mp[i] = Bb * xv + D * pv;
    }
#pragma unroll
    for (int w = 0; w < NMODES - 1; ++w) {
      float ct = s_ct[cc][l][w], cs = s_cs[cc][l][w], ss = s_ss[cc][l][w];
      float x1 = mx[w], x2 = mx[w + 1], p1 = mp[w], p2 = mp[w + 1];
      mx[w]     = ct * x1 - cs * x2 - ss * p2;
      mx[w + 1] = cs * x1 + ct * x2 - ss * p1;
      mp[w]     = ss * x2 + ct * p1 - cs * p2;
      mp[w + 1] = ss * x1 + cs * p1 + ct * p2;
    }
  }

  // Circuit 0 contributes output columns 0..15 (mx); circuit 1 -> 16..31 (mp).
  if (col < 16) {
#pragma unroll
    for (int j = 0; j < NMODES; ++j)
      ws[col * 32 + 16 * cc + j] = (cc == 0) ? mx[j] : mp[j];
  } else {
#pragma unroll
    for (int j = 0; j < NMODES; ++j)
      ws[512 + 16 * cc + j] = (cc == 0) ? mx[j] : mp[j];
  }
}

// ---------------------------------------------------------------------------
// Main GEMM: out[B,32] = x[B,16] * Wt[16,32] + bias, via WMMA f32 16x16x4.
// One wave handles a 16-sample tile: 4 K-chunks x 2 output halves = 8 WMMAs.
// ---------------------------------------------------------------------------
__global__ __launch_bounds__(256) void cv_gemm_kernel(
    const float* __restrict__ x, const float* __restrict__ ws,
    float* __restrict__ out, int numTiles)
{
  const int lane = threadIdx.x & 31;
  const int row  = lane & 15;   // M (or N) index within half-wave
  const int hi   = lane >> 4;   // which half of the wave

  // B fragments (Wt), kept in registers across all tiles.
  // 4x16 f32 B layout: VGPR0 = rows K=0 (lanes 0-15) / K=2 (lanes 16-31),
  //                    VGPR1 = rows K=1 / K=3.   B[k][n] = Wt[4*kc+k][16*h+n]
  v2f bfrag[2][4];
#pragma unroll
  for (int h = 0; h < 2; ++h)
#pragma unroll
    for (int kc = 0; kc < 4; ++kc) {
      bfrag[h][kc].x = ws[(4 * kc + 0 + 2 * hi) * 32 + 16 * h + row];
      bfrag[h][kc].y = ws[(4 * kc + 1 + 2 * hi) * 32 + 16 * h + row];
    }
  const float bias0 = ws[512 + row];        // N = lane&15 for every D VGPR
  const float bias1 = ws[512 + 16 + row];

  const int waveId     = blockIdx.x * 8 + (threadIdx.x >> 5);
  const int totalWaves = gridDim.x * 8;

  for (int tile = waveId; tile < numTiles; tile += totalWaves) {
    const float* xt = x + (size_t)tile * (16 * 16);

    // A fragments: 16x4 f32 A layout.  Lane (hi,row) holds x[row][4kc+2hi+{0,1}].
    v2f afrag[4];
#pragma unroll
    for (int kc = 0; kc < 4; ++kc) {
      const v2f* p = (const v2f*)(xt + row * 16 + 4 * kc + 2 * hi);
      afrag[kc] = __builtin_nontemporal_load(p);
    }

    v8f acc0 = {};  // output columns 0..15  (circuit 0 mx)
    v8f acc1 = {};  // output columns 16..31 (circuit 1 mp)
#pragma unroll
    for (int kc = 0; kc < 4; ++kc) {
      acc0 = __builtin_amdgcn_wmma_f32_16x16x4_f32(
          false, afrag[kc], false, bfrag[0][kc], (short)0, acc0, false, false);
      acc1 = __builtin_amdgcn_wmma_f32_16x16x4_f32(
          false, afrag[kc], false, bfrag[1][kc], (short)0, acc1, false, false);
    }

    // D layout: VGPR v -> row (v + 8*hi), column = row(lane&15) within half.
    float* ot = out + (size_t)tile * (16 * 32);
#pragma unroll
    for (int v = 0; v < 8; ++v) {
      int m = v + 8 * hi;
      __builtin_nontemporal_store(acc0[v] + bias0, &ot[m * 32 + row]);
      __builtin_nontemporal_store(acc1[v] + bias1, &ot[m * 32 + 16 + row]);
    }
  }
}

extern "C" void kernel_launch(void* const* d_in, const int* in_sizes, int n_in,
                              void* d_out, int out_size, void* d_ws, size_t ws_size,
                              hipStream_t stream) {
  const float* x    = (const float*)d_in[0];
  const float* disp = (const float*)d_in[1];
  const float* sq   = (const float*)d_in[2];
  const float* bs   = (const float*)d_in[3];
  float* out = (float*)d_out;
  float* ws  = (float*)d_ws;

  const int BN       = in_sizes[0];   // B * 16
  const int numTiles = BN / 256;      // B / 16 samples per wave-tile

  hipLaunchKernelGGL(cv_setup_kernel, dim3(1), dim3(192), 0, stream,
                     disp, sq, bs, ws);

  int blocks = (numTiles + 31) / 32;          // ~4 tiles per wave
  if (blocks < 1) blocks = 1;
  if (blocks > 4096) blocks = 4096;
  hipLaunchKernelGGL(cv_gemm_kernel, dim3(blocks), dim3(256), 0, stream,
                     x, ws, out, numTiles);
}